// QLSTMCell_65481071399076
// MI455X (gfx1250) — compile-verified
//
#include <hip/hip_runtime.h>
#include <math.h>

typedef float v2f __attribute__((ext_vector_type(2)));
typedef float v4f __attribute__((ext_vector_type(4)));
typedef float v8f __attribute__((ext_vector_type(8)));

#define BATCH 32768
#define DIN   4096
#define KHALF 2048                 // split-K: two independent WMMA chains
#define H     4
#define DTOT  4100                 // DIN + H
#define NCOL  16                   // 4 gates * 4 NQ columns
#define WAVES 8
#define ROWS_PER_WAVE 16
#define ROWS_PER_BLOCK (WAVES * ROWS_PER_WAVE)   // 128

#define LDS_W_FLOATS   (NCOL * DTOT)             // 65600 floats = 262400 B
#define SCR_STRIDE     17                        // pad 16x16 tile rows -> conflict-free
#define LDS_SCR_FLOATS (WAVES * ROWS_PER_WAVE * SCR_STRIDE)   // 2176 floats
#define LDS_TOTAL_BYTES ((LDS_W_FLOATS + LDS_SCR_FLOATS) * 4) // 271104 B < 320KB

#define PF_DIST 1024               // prefetch distance in floats (4KB ahead per row)

__device__ __forceinline__ float sigmoidf_(float v) {
    return 1.0f / (1.0f + expf(-v));
}

__global__ void __launch_bounds__(256, 1)
qlstm_wmma_kernel(const float* __restrict__ x,  const float* __restrict__ hx,
                  const float* __restrict__ cx,
                  const float* __restrict__ Wf, const float* __restrict__ bf, const float* __restrict__ thf,
                  const float* __restrict__ Wi, const float* __restrict__ bi, const float* __restrict__ thi,
                  const float* __restrict__ Wg, const float* __restrict__ bg, const float* __restrict__ thg,
                  const float* __restrict__ Wo, const float* __restrict__ bo, const float* __restrict__ tho,
                  float* __restrict__ out)
{
    extern __shared__ float smem[];

    const int tid  = threadIdx.x;
    const int wave = tid >> 5;          // wave32
    const int lane = tid & 31;

    // ---------------- Stage weights into LDS (b128 copies) ----------------
    // LDS row n (0..15): gate = n/4 (f,i,g,o), q = n%4, full 4100 columns.
    {
        const float* Wptr[4] = { Wf, Wi, Wg, Wo };
        #pragma unroll
        for (int n = 0; n < NCOL; ++n) {
            const v4f* src = (const v4f*)(Wptr[n >> 2] + (n & 3) * DTOT);
            v4f* dst = (v4f*)(smem + n * DTOT);
            for (int q = tid; q < DTOT / 4; q += 256) dst[q] = src[q];   // 1025 v4f per row
        }
    }
    __syncthreads();

    // ---------------- WMMA accumulation over K (split into 2 chains) ------
    // A (16x4 f32) layout: lane&15 = M; VGPR0/1 hold K = k0+off, k0+off+1
    // with off = 0 (lanes 0-15) or 2 (lanes 16-31) -> one b64 load per lane.
    const int mrow = lane & 15;
    const int koff = (lane >> 4) << 1;  // 0 or 2
    const int rowBase = blockIdx.x * ROWS_PER_BLOCK + wave * ROWS_PER_WAVE;
    const int arow = rowBase + mrow;

    const float* aptr = x + (size_t)arow * DIN + koff;              // global A stream
    const float* wlds = smem + mrow * DTOT + koff;                  // LDS B stream (conflict-free)

    v8f acc0 = {0.f, 0.f, 0.f, 0.f, 0.f, 0.f, 0.f, 0.f};
    v8f acc1 = {0.f, 0.f, 0.f, 0.f, 0.f, 0.f, 0.f, 0.f};

    #pragma unroll 8
    for (int k0 = 0; k0 < KHALF; k0 += 4) {
        if ((k0 & 31) == 0) {   // once per 128B line of stream advance (folded at compile time)
            __builtin_prefetch(aptr + k0 + PF_DIST, 0, 1);
            __builtin_prefetch(aptr + KHALF + k0 + PF_DIST, 0, 1);
        }
        v2f a0 = *(const v2f*)(aptr + k0);
        v2f b0 = *(const v2f*)(wlds + k0);
        v2f a1 = *(const v2f*)(aptr + KHALF + k0);
        v2f b1 = *(const v2f*)(wlds + KHALF + k0);
        acc0 = __builtin_amdgcn_wmma_f32_16x16x4_f32(
                   false, a0, false, b0, (short)0, acc0, false, false);
        acc1 = __builtin_amdgcn_wmma_f32_16x16x4_f32(
                   false, a1, false, b1, (short)0, acc1, false, false);
    }
    // Tail K-step: columns 4096..4099 come from hx
    {
        v2f a = *(const v2f*)(hx + (size_t)arow * H + koff);
        v2f b = *(const v2f*)(wlds + DIN);
        acc0 = __builtin_amdgcn_wmma_f32_16x16x4_f32(
                   false, a, false, b, (short)0, acc0, false, false);
    }
    v8f acc = acc0 + acc1;

    // ---------------- Spill 16x16 tile to padded LDS scratch ----------------
    float* scr = smem + LDS_W_FLOATS + wave * (ROWS_PER_WAVE * SCR_STRIDE);
    const int rhalf = (lane >> 4) * 8;  // C/D layout: VGPR v -> M = v (+8 for lanes 16-31)
    #pragma unroll
    for (int v = 0; v < 8; ++v)
        scr[(v + rhalf) * SCR_STRIDE + mrow] = acc[v];
    __syncthreads();

    // ---------------- Per-row quantum-gate epilogue ----------------
    if (lane < 16) {
        const int grow = rowBase + lane;

        float z[16];
        #pragma unroll
        for (int n = 0; n < 16; ++n) z[n] = scr[lane * SCR_STRIDE + n];

        const float* bArr[4] = { bf, bi, bg, bo };
        const float* tArr[4] = { thf, thi, thg, tho };

        float act[4][4];
        #pragma unroll
        for (int gi = 0; gi < 4; ++gi) {
            float cc[4];
            #pragma unroll
            for (int q = 0; q < 4; ++q)
                cc[q] = __cosf(z[gi * 4 + q] + bArr[gi][q] + tArr[gi][q]);
            float t12 = cc[1] * cc[2];
            float z0 = t12 * cc[3];
            float z1 = cc[0] * cc[1];
            float z2 = cc[0] * t12;
            float z3 = z2 * cc[3];
            float zz[4] = { z0, z1, z2, z3 };
            #pragma unroll
            for (int q = 0; q < 4; ++q)
                act[gi][q] = (gi == 2) ? tanhf(zz[q]) : sigmoidf_(zz[q]);
        }

        v4f cxv = *(const v4f*)(cx + (size_t)grow * 4);
        v4f cnew, hnew;
        #pragma unroll
        for (int q = 0; q < 4; ++q) {
            float cn = act[0][q] * cxv[q] + act[1][q] * act[2][q];  // f*cx + i*g
            cnew[q] = cn;
            hnew[q] = act[3][q] * tanhf(cn);                        // o * tanh(c)
        }
        *(v4f*)(out + (size_t)grow * 4) = hnew;                      // hx_new
        *(v4f*)(out + (size_t)BATCH * 4 + (size_t)grow * 4) = cnew;  // cx_new
    }
}

extern "C" void kernel_launch(void* const* d_in, const int* in_sizes, int n_in,
                              void* d_out, int out_size, void* d_ws, size_t ws_size,
                              hipStream_t stream) {
    (void)in_sizes; (void)n_in; (void)out_size; (void)d_ws; (void)ws_size;
    const float* x   = (const float*)d_in[0];
    const float* hx  = (const float*)d_in[1];
    const float* cx  = (const float*)d_in[2];
    const float* Wf  = (const float*)d_in[3];
    const float* bf  = (const float*)d_in[4];
    const float* thf = (const float*)d_in[5];
    const float* Wi  = (const float*)d_in[6];
    const float* bi  = (const float*)d_in[7];
    const float* thi = (const float*)d_in[8];
    const float* Wg  = (const float*)d_in[9];
    const float* bg  = (const float*)d_in[10];
    const float* thg = (const float*)d_in[11];
    const float* Wo  = (const float*)d_in[12];
    const float* bo  = (const float*)d_in[13];
    const float* tho = (const float*)d_in[14];

    dim3 grid(BATCH / ROWS_PER_BLOCK);   // 256
    dim3 block(256);                     // 8 wave32 waves
    hipLaunchKernelGGL(qlstm_wmma_kernel, grid, block, LDS_TOTAL_BYTES, stream,
                       x, hx, cx, Wf, bf, thf, Wi, bi, thi, Wg, bg, thg,
                       Wo, bo, tho, (float*)d_out);
}